// ClassCaps_35656818491745
// MI455X (gfx1250) — compile-verified
//
#include <hip/hip_runtime.h>
#include <math.h>
#include <stdint.h>

// ---------------------------------------------------------------------------
// EM-routing capsule layer for MI455X (gfx1250), wave32, fp32 WMMA 16x16x4.
//
// Decomposition: inc = 64 spatial (s) x 32 channels (c).
//   votes[b,s,c,j,d] = W[c,j,d]*pose[b,s,c,d] + coord(s,d)   (coord: d=0,1)
//
// Feature matrix per (b,c):  F[s,q], q in 0..47 (39 real + pad)
//   q 0..15 : p_d          q16: c0   q17: c1   q18: 1
//   q19..34 : p_d^2        q35: p0*c0  q36: p1*c1  q37: c0^2  q38: c1^2
//
// m-step GEMM (per b,c):  G[j,q] = sum_s (r[s,j]*act[s]) * F[s,q]
//   -> r_sum, sum r*v, sum r*v^2 all reconstructed from G with W.
// e-step GEMM (per b,c):  cost[s,j] = sum_q F[s,q] * g[q,j]
//   -> full Gaussian log-prob expanded into coefficients g over the same F.
//
// Pose tiles (64x16 f32 = 4KB per (b,c)) are staged into LDS with
// GLOBAL_LOAD_ASYNC_TO_LDS_B128 (ASYNCcnt) before building F.
// ---------------------------------------------------------------------------

#define B_  32
#define S_  64
#define C_  32
#define J_  100
#define JP_ 112      // J padded to 7 WMMA N-tiles
#define D_  16
#define Q_  48       // feature cols padded to 3 WMMA tiles / 12 K-steps
#define EPS_     1e-7f
#define VAR_EPS_ 1e-2f

typedef __attribute__((ext_vector_type(2))) float v2f;
typedef __attribute__((ext_vector_type(8))) float v8f;

__device__ __forceinline__ float wave_sum32(float v) {
  #pragma unroll
  for (int off = 16; off > 0; off >>= 1) v += __shfl_down(v, off, 32);
  return __shfl(v, 0, 32);
}

// Async-copy the 64x16 f32 pose tile for (b,c) into LDS.
// 256 threads x 16B = 4KB, one global_load_async_to_lds_b128 per thread.
__device__ __forceinline__ void stage_pose_async(float (*P)[D_],
                                                 const float* __restrict__ pose,
                                                 int b, int c) {
  int t = threadIdx.x;               // blockDim.x == 256
  int s = t >> 2, part = (t & 3) * 4;
  const float* gp = pose + (((size_t)b * S_ + s) * C_ + c) * D_ + part;
  uint32_t lds_addr = (uint32_t)(uintptr_t)(&P[s][part]);
  asm volatile("global_load_async_to_lds_b128 %0, %1, off"
               :: "v"(lds_addr), "v"(gp) : "memory");
  asm volatile("s_wait_asynccnt 0x0" ::: "memory");
}

// Build the 64x48 feature matrix from the LDS pose tile.
__device__ __forceinline__ void build_F(float (*F)[Q_], const float (*P)[D_]) {
  for (int idx = threadIdx.x; idx < S_ * Q_; idx += blockDim.x) {
    int s = idx / Q_, q = idx % Q_;
    float c0 = ((float)(s >> 3) + 0.5f) * 0.125f;   // (y+0.5)/8
    float c1 = ((float)(s & 7) + 0.5f) * 0.125f;    // (x+0.5)/8
    float v;
    if (q < 16)        v = P[s][q];
    else if (q == 16)  v = c0;
    else if (q == 17)  v = c1;
    else if (q == 18)  v = 1.0f;
    else if (q < 35) { float t = P[s][q - 19]; v = t * t; }
    else if (q == 35)  v = P[s][0] * c0;
    else if (q == 36)  v = P[s][1] * c1;
    else if (q == 37)  v = c0 * c0;
    else if (q == 38)  v = c1 * c1;
    else               v = 0.0f;
    F[s][q] = v;
  }
}

__global__ void k_init_r(float* __restrict__ r) {
  size_t n = (size_t)B_ * C_ * S_ * JP_;
  for (size_t i = (size_t)blockIdx.x * blockDim.x + threadIdx.x; i < n;
       i += (size_t)gridDim.x * blockDim.x) {
    int j = (int)(i % JP_);
    r[i] = (j < J_) ? (1.0f / (float)J_) : 0.0f;
  }
}

// ---------------- m-step moment GEMM: G[j,q] = sum_s rw[s,j] * F[s,q] -------
__global__ __launch_bounds__(256)
void k_mstep_gemm(const float* __restrict__ pose, const float* __restrict__ active,
                  const float* __restrict__ r, float* __restrict__ G) {
  __shared__ __align__(16) float P[S_][D_];   // 4 KB pose tile
  __shared__ float F[S_][Q_];                 // 12 KB features
  int blk = blockIdx.x;             // b*C + c
  int b = blk / C_, c = blk % C_;
  stage_pose_async(P, pose, b, c);
  __syncthreads();
  build_F(F, P);
  __syncthreads();

  int wave = threadIdx.x >> 5;      // 8 waves
  int lane = threadIdx.x & 31;
  int half = lane >> 4, lm = lane & 15;
  const float* rblk = r + (size_t)blk * S_ * JP_;
  float* Gp = G + (size_t)blk * JP_ * Q_;

  // 7 M-tiles (j) x 3 N-tiles (q) = 21 tiles, K = 64 (s) in 16 steps of 4
  for (int t = wave; t < 7 * 3; t += 8) {
    int mbase = (t / 3) * 16;
    int nbase = (t % 3) * 16;
    int j = mbase + lm;
    v8f acc = {};
    #pragma unroll 4
    for (int kb = 0; kb < 16; ++kb) {
      int s0 = kb * 4 + half * 2;
      float a0 = rblk[(size_t)s0 * JP_ + j] *
                 active[((size_t)b * S_ + s0) * C_ + c];
      float a1 = rblk[(size_t)(s0 + 1) * JP_ + j] *
                 active[((size_t)b * S_ + (s0 + 1)) * C_ + c];
      v2f av = {a0, a1};
      v2f bv_ = {F[s0][nbase + lm], F[s0 + 1][nbase + lm]};
      acc = __builtin_amdgcn_wmma_f32_16x16x4_f32(false, av, false, bv_,
                                                  (short)0, acc, false, false);
    }
    #pragma unroll
    for (int i = 0; i < 8; ++i) {
      int row = mbase + (half ? 8 : 0) + i;
      Gp[(size_t)row * Q_ + nbase + lm] = acc[i];
    }
  }
}

// ---------------- m-step stats: mu, sigma, act_p per (b, j) ----------------
__global__ void k_mstep_stats(const float* __restrict__ G, const float* __restrict__ W,
                              const float* __restrict__ bv, const float* __restrict__ ba,
                              float* __restrict__ mu, float* __restrict__ e2,
                              float* __restrict__ cst, float* __restrict__ out,
                              float lam, int last) {
  int b = blockIdx.x / J_;
  int j = blockIdx.x % J_;
  int c = threadIdx.x;              // 32 lanes = 32 channels
  const float* g = G + (((size_t)(b * C_ + c)) * JP_ + j) * Q_;

  float rsum = wave_sum32(g[18]);
  float inv_rs = 1.0f / (rsum + EPS_);
  float logsum = 0.0f;              // sum_d log(sigma + EPS)
  float emusum = 0.0f;              // sum_d e * mu^2
  float mu_val = 0.0f, e_val = 0.0f;

  for (int d = 0; d < D_; ++d) {
    float w = W[((size_t)c * J_ + j) * D_ + d];
    float s1p  = w * g[d];
    float sv2p = w * w * g[19 + d];
    if (d == 0) { s1p += g[16]; sv2p += 2.0f * w * g[35] + g[37]; }
    if (d == 1) { s1p += g[17]; sv2p += 2.0f * w * g[36] + g[38]; }
    float S1  = wave_sum32(s1p);
    float Sv2 = wave_sum32(sv2p);
    float m   = S1 * inv_rs;
    float var = (Sv2 - 2.0f * m * S1 + m * m * rsum) * inv_rs + VAR_EPS_;
    var = fmaxf(var, 1e-6f);
    float sig = sqrtf(var);
    float e   = 1.0f / (2.0f * var + EPS_);
    logsum += logf(sig + EPS_);
    emusum += e * m * m;
    if (c == d) { mu_val = m; e_val = e; }
  }

  float lh = (16.0f * bv[j] + logsum) * rsum;           // sum_d (bv+log(sig+EPS))*rsum
  float ap = 1.0f / (1.0f + expf(-lam * (ba[j] - lh))); // sigmoid
  float p0 = -logsum;

  if (c < D_) {
    mu[((size_t)b * J_ + j) * D_ + c] = mu_val;
    e2[((size_t)b * J_ + j) * D_ + c] = e_val;
  }
  if (c == 0) cst[(size_t)b * J_ + j] = logf(ap + EPS_) + p0 - emusum;
  if (last) {
    if (c < D_) out[(size_t)b * 1600 + j * D_ + c] = mu_val;
    if (c == 0) out[(size_t)B_ * 1600 + (size_t)b * J_ + j] = ap;
  }
}

// ---------------- e-step: cost GEMM + row softmax -> new r ------------------
__global__ __launch_bounds__(256)
void k_estep(const float* __restrict__ pose, const float* __restrict__ W,
             const float* __restrict__ mu, const float* __restrict__ e2,
             const float* __restrict__ cst, float* __restrict__ r) {
  __shared__ __align__(16) float P[S_][D_];   // 4 KB pose tile
  __shared__ float F[S_][Q_];                 // 12 KB
  __shared__ float Gg[Q_][JP_];               // 21 KB coefficients g[q][j]
  __shared__ float cost[S_][JP_];             // 28 KB
  int blk = blockIdx.x;
  int b = blk / C_, c = blk % C_;
  stage_pose_async(P, pose, b, c);

  // Build coefficient matrix g[q][j]: cost = logact + p0 - sum_d e*(w*p+coord-mu)^2
  for (int idx = threadIdx.x; idx < Q_ * JP_; idx += blockDim.x) {
    int q = idx / JP_, j = idx % JP_;
    float v = 0.0f;
    if (j < J_) {
      const float* mp = mu + ((size_t)b * J_ + j) * D_;
      const float* ep = e2 + ((size_t)b * J_ + j) * D_;
      const float* wp = W + ((size_t)c * J_ + j) * D_;
      if (q < 16)        v = 2.0f * ep[q] * wp[q] * mp[q];
      else if (q == 16)  v = 2.0f * ep[0] * mp[0];
      else if (q == 17)  v = 2.0f * ep[1] * mp[1];
      else if (q == 18)  v = cst[(size_t)b * J_ + j];
      else if (q < 35) { int d = q - 19; v = -ep[d] * wp[d] * wp[d]; }
      else if (q == 35)  v = -2.0f * ep[0] * wp[0];
      else if (q == 36)  v = -2.0f * ep[1] * wp[1];
      else if (q == 37)  v = -ep[0];
      else if (q == 38)  v = -ep[1];
    } else {
      v = (q == 18) ? -1e30f : 0.0f;  // padded classes -> -inf logits
    }
    Gg[q][j] = v;
  }
  __syncthreads();
  build_F(F, P);
  __syncthreads();

  int wave = threadIdx.x >> 5, lane = threadIdx.x & 31;
  int half = lane >> 4, lm = lane & 15;
  // 4 M-tiles (s) x 7 N-tiles (j) = 28 tiles, K = 48 in 12 steps of 4
  for (int t = wave; t < 4 * 7; t += 8) {
    int mbase = (t / 7) * 16;
    int nbase = (t % 7) * 16;
    v8f acc = {};
    #pragma unroll 4
    for (int kb = 0; kb < 12; ++kb) {
      int k0 = kb * 4 + half * 2;
      v2f av = {F[mbase + lm][k0], F[mbase + lm][k0 + 1]};
      v2f bv_ = {Gg[k0][nbase + lm], Gg[k0 + 1][nbase + lm]};
      acc = __builtin_amdgcn_wmma_f32_16x16x4_f32(false, av, false, bv_,
                                                  (short)0, acc, false, false);
    }
    #pragma unroll
    for (int i = 0; i < 8; ++i)
      cost[mbase + (half ? 8 : 0) + i][nbase + lm] = acc[i];
  }
  __syncthreads();

  // Softmax over classes per (s) row; write new r (padded classes = 0).
  for (int s = threadIdx.x; s < S_; s += blockDim.x) {
    float mx = -3.4e38f;
    for (int j = 0; j < J_; ++j) mx = fmaxf(mx, cost[s][j]);
    float sum = 0.0f;
    for (int j = 0; j < J_; ++j) sum += expf(cost[s][j] - mx);
    float inv = 1.0f / sum;
    float* rp = r + ((size_t)blk * S_ + s) * JP_;
    for (int j = 0; j < J_; ++j) rp[j] = expf(cost[s][j] - mx) * inv;
    for (int j = J_; j < JP_; ++j) rp[j] = 0.0f;
  }
}

// ---------------------------------------------------------------------------
extern "C" void kernel_launch(void* const* d_in, const int* in_sizes, int n_in,
                              void* d_out, int out_size, void* d_ws, size_t ws_size,
                              hipStream_t stream) {
  (void)in_sizes; (void)n_in; (void)out_size; (void)ws_size;
  const float* pose   = (const float*)d_in[0];
  const float* active = (const float*)d_in[1];
  const float* W      = (const float*)d_in[2];
  const float* bv     = (const float*)d_in[3];
  const float* ba     = (const float*)d_in[4];
  float* out = (float*)d_out;

  float* ws  = (float*)d_ws;
  float* r   = ws;                                  // B*C*S*JP   = 7,340,032
  float* G   = r  + (size_t)B_ * C_ * S_ * JP_;     // B*C*JP*Q   = 5,505,024
  float* mu  = G  + (size_t)B_ * C_ * JP_ * Q_;     // B*J*D      = 51,200
  float* e2  = mu + (size_t)B_ * J_ * D_;           // B*J*D      = 51,200
  float* cst = e2 + (size_t)B_ * J_ * D_;           // B*J        = 3,200

  k_init_r<<<256, 256, 0, stream>>>(r);
  for (int it = 0; it < 3; ++it) {
    float lam = 0.01f * (1.0f - powf(0.95f, (float)(it + 1)));
    k_mstep_gemm<<<B_ * C_, 256, 0, stream>>>(pose, active, r, G);
    k_mstep_stats<<<B_ * J_, 32, 0, stream>>>(G, W, bv, ba, mu, e2, cst, out,
                                              lam, (it == 2) ? 1 : 0);
    if (it < 2) k_estep<<<B_ * C_, 256, 0, stream>>>(pose, W, mu, e2, cst, r);
  }
}